// GCN_45148696216046
// MI455X (gfx1250) — compile-verified
//
#include <hip/hip_runtime.h>

// ---------------------------------------------------------------------------
// 2-layer GCN, algebraically refactored:
//   out1 = S (X W1) + b1 = (S X) W1 + b1            (S = D^-1/2 (A+I) D^-1/2)
//   z    = relu(out1) @ W2                          (scalar per node)
//   out  = S z + b2
// Edge work operates on 2-wide (layer 1) / 1-wide (layer 2) payloads only.
// Dense stage uses V_WMMA_F32_16X16X4_F32 on gfx1250 (wave32), branchless.
// ---------------------------------------------------------------------------

typedef float v2f __attribute__((ext_vector_type(2)));
typedef float v8f __attribute__((ext_vector_type(8)));

// ---- init: deg=1 (self loop), agg=0, z=0, out=0 ---------------------------
__global__ void gcn_init(float* __restrict__ deg, float* __restrict__ agg,
                         float* __restrict__ z, float* __restrict__ out, int n) {
  int i = blockIdx.x * blockDim.x + threadIdx.x;
  if (i < n) {
    deg[i] = 1.0f;
    agg[2 * i] = 0.0f;
    agg[2 * i + 1] = 0.0f;
    z[i] = 0.0f;
    out[i] = 0.0f;
  }
}

// ---- in-degree accumulation over edges ------------------------------------
__global__ void gcn_degree(const int* __restrict__ dst, float* __restrict__ deg,
                           int E) {
  int i = blockIdx.x * blockDim.x + threadIdx.x;
  int stride = gridDim.x * blockDim.x;
  for (; i < E; i += stride) {
    if (i + stride < E) __builtin_prefetch(&dst[i + stride], 0, 1);
    atomicAdd(&deg[dst[i]], 1.0f);
  }
}

// ---- dinv = rsqrt(deg), in place ------------------------------------------
__global__ void gcn_rsqrt(float* __restrict__ dinv, int n) {
  int i = blockIdx.x * blockDim.x + threadIdx.x;
  if (i < n) dinv[i] = rsqrtf(dinv[i]);
}

// ---- layer-1 edge aggregation of the RAW 2-wide features ------------------
// agg[d] += x[s] * dinv[s] * dinv[d]
__global__ void gcn_agg2(const int* __restrict__ src, const int* __restrict__ dst,
                         const float* __restrict__ x, const float* __restrict__ dinv,
                         float* __restrict__ agg, int E) {
  int i = blockIdx.x * blockDim.x + threadIdx.x;
  int stride = gridDim.x * blockDim.x;
  for (; i < E; i += stride) {
    if (i + stride < E) {
      __builtin_prefetch(&src[i + stride], 0, 1);
      __builtin_prefetch(&dst[i + stride], 0, 1);
    }
    int s = src[i];
    int d = dst[i];
    float w = dinv[s] * dinv[d];
    atomicAdd(&agg[2 * d], x[2 * s] * w);
    atomicAdd(&agg[2 * d + 1], x[2 * s + 1] * w);
  }
}

// ---- fused dense stage: z = relu((S X) W1 + b1) @ W2 ----------------------
// One wave handles 16 nodes. A (16x4 f32): lanes 0-15 carry K=0,1 (the two
// real features), lanes 16-31 carry K=2,3 == 0 padding. B (4x16): mirrored
// layout, W1 rows in lanes 0-15. C/D (16x16 f32, 8 VGPRs): lane = column N,
// VGPR v = row M=v (lanes 0-15) / M=v+8 (lanes 16-31). b1 folded into C.
// All loads are unconditional (addresses valid for every lane); K/row padding
// is applied with a 0/1 multiplicative mask so EXEC stays all-ones and the
// inner loop is straight-line (no exec save/restore per WMMA).
__global__ __launch_bounds__(256) void gcn_dense_wmma(
    const float* __restrict__ agg, const float* __restrict__ x,
    const float* __restrict__ dinv, const float* __restrict__ W1,
    const float* __restrict__ b1, const float* __restrict__ W2,
    float* __restrict__ z, int n) {
  const int lane = threadIdx.x & 31;
  const int wave = threadIdx.x >> 5;
  const int tile = blockIdx.x * 8 + wave;
  const int base = tile * 16;
  if (base >= n) return;  // uniform across the wave -> EXEC stays all-ones

  const int m = lane & 15;
  const bool hiK = lane >= 16;          // lanes 16-31 supply the K=2,3 zeros
  const float kmask = hiK ? 0.0f : 1.0f;

  // A operand: aggregated features + self-loop term, rows base..base+15.
  // Clamp the node index so the load is always in-range, then mask.
  int node = base + m;
  const float rowmask = (!hiK && node < n) ? 1.0f : 0.0f;
  if (node >= n) node = n - 1;
  {
    // nothing
  }
  float di = dinv[node];
  float sl = di * di;
  v2f A;
  A.x = (agg[2 * node]     + x[2 * node]     * sl) * rowmask;
  A.y = (agg[2 * node + 1] + x[2 * node + 1] * sl) * rowmask;

  float zacc[8];
#pragma unroll
  for (int v = 0; v < 8; ++v) zacc[v] = 0.0f;

#pragma unroll
  for (int n0 = 0; n0 < 128; n0 += 16) {
    int col = n0 + m;                 // valid for all 32 lanes
    v2f B;
    B.x = W1[col]       * kmask;      // W1 [2,128] row-major
    B.y = W1[128 + col] * kmask;
    float bias = b1[col];
    v8f C;
#pragma unroll
    for (int v = 0; v < 8; ++v) C[v] = bias;

    // D = A x B + C   (16x16x4 f32 WMMA)
    C = __builtin_amdgcn_wmma_f32_16x16x4_f32(
        /*neg_a=*/false, A, /*neg_b=*/false, B,
        /*c_mod=*/(short)0, C, /*reuse_a=*/false, /*reuse_b=*/false);

    float w2 = W2[col];
#pragma unroll
    for (int v = 0; v < 8; ++v) {
      float h = C[v];
      h = h > 0.0f ? h : 0.0f;  // ReLU
      zacc[v] += h * w2;        // partial of row-dot with W2
    }
  }

  // Reduce the row-dot across the 16 lanes of each half-wave.
#pragma unroll
  for (int v = 0; v < 8; ++v) {
    float r = zacc[v];
    r += __shfl_xor(r, 1, 16);
    r += __shfl_xor(r, 2, 16);
    r += __shfl_xor(r, 4, 16);
    r += __shfl_xor(r, 8, 16);
    zacc[v] = r;
  }
  if (m == 0) {  // lane 0 writes rows 0-7, lane 16 writes rows 8-15
    int rowbase = base + (hiK ? 8 : 0);
#pragma unroll
    for (int v = 0; v < 8; ++v) {
      int r = rowbase + v;
      if (r < n) z[r] = zacc[v];
    }
  }
}

// ---- layer-2 self-loop + bias: out[i] = z[i]*dinv[i]^2 + b2 ---------------
__global__ void gcn_out_self(const float* __restrict__ z,
                             const float* __restrict__ dinv,
                             const float* __restrict__ b2,
                             float* __restrict__ out, int n) {
  int i = blockIdx.x * blockDim.x + threadIdx.x;
  if (i < n) out[i] = z[i] * dinv[i] * dinv[i] + b2[0];
}

// ---- layer-2 scalar edge aggregation --------------------------------------
__global__ void gcn_out_edges(const int* __restrict__ src, const int* __restrict__ dst,
                              const float* __restrict__ z,
                              const float* __restrict__ dinv,
                              float* __restrict__ out, int E) {
  int i = blockIdx.x * blockDim.x + threadIdx.x;
  int stride = gridDim.x * blockDim.x;
  for (; i < E; i += stride) {
    if (i + stride < E) {
      __builtin_prefetch(&src[i + stride], 0, 1);
      __builtin_prefetch(&dst[i + stride], 0, 1);
    }
    int s = src[i];
    int d = dst[i];
    atomicAdd(&out[d], z[s] * dinv[s] * dinv[d]);
  }
}

extern "C" void kernel_launch(void* const* d_in, const int* in_sizes, int n_in,
                              void* d_out, int out_size, void* d_ws, size_t ws_size,
                              hipStream_t stream) {
  const float* x  = (const float*)d_in[0];   // [n,2]
  const int*   ei = (const int*)d_in[1];     // [2,E] row-major: src then dst
  const float* W1 = (const float*)d_in[2];   // [2,128]
  const float* b1 = (const float*)d_in[3];   // [128]
  const float* W2 = (const float*)d_in[4];   // [128,1]
  const float* b2 = (const float*)d_in[5];   // [1]

  int n = in_sizes[0] / 2;
  int E = in_sizes[1] / 2;
  const int* src = ei;
  const int* dst = ei + E;
  float* out = (float*)d_out;

  // Workspace: dinv[n] | agg[2n] | z[n]  (4n floats total)
  float* dinv = (float*)d_ws;
  float* agg  = dinv + n;
  float* z    = agg + 2 * (size_t)n;

  const int tb = 256;
  int gN = (n + tb - 1) / tb;
  int gE = (E + tb - 1) / tb;
  if (gE > 4096) gE = 4096;  // grid-stride edge kernels

  gcn_init<<<gN, tb, 0, stream>>>(dinv, agg, z, out, n);
  gcn_degree<<<gE, tb, 0, stream>>>(dst, dinv, E);
  gcn_rsqrt<<<gN, tb, 0, stream>>>(dinv, n);
  gcn_agg2<<<gE, tb, 0, stream>>>(src, dst, x, dinv, agg, E);

  int tiles = (n + 15) / 16;                 // 16 nodes per wave
  int dblocks = (tiles + 7) / 8;             // 8 waves per 256-thread block
  gcn_dense_wmma<<<dblocks, 256, 0, stream>>>(agg, x, dinv, W1, b1, W2, z, n);

  gcn_out_self<<<gN, tb, 0, stream>>>(z, dinv, b2, out, n);
  gcn_out_edges<<<gE, tb, 0, stream>>>(src, dst, z, dinv, out, E);
}